// ConformanceGNN_29403346108947
// MI455X (gfx1250) — compile-verified
//
#include <hip/hip_runtime.h>
#include <hip/hip_bf16.h>

#define HID 256

typedef __attribute__((ext_vector_type(2))) float v2f;
typedef __attribute__((ext_vector_type(8))) float v8f;

// -------------------- encode: out[i,h] = x[i]*W[h] + b[h] --------------------
__global__ __launch_bounds__(256)
void encode_kernel(const float* __restrict__ x, const float* __restrict__ W,
                   const float* __restrict__ b, float* __restrict__ out, int N) {
    size_t i = (size_t)blockIdx.x * 256 + threadIdx.x;
    if (i >= (size_t)N * HID) return;
    int node = (int)(i >> 8);
    int h    = (int)(i & 255);
    out[i] = x[node] * W[h] + b[h];
}

// -------------------- degree count (float) --------------------
__global__ __launch_bounds__(256)
void count_kernel(const int* __restrict__ dst, float* __restrict__ cnt, int E) {
    int e = blockIdx.x * 256 + threadIdx.x;
    if (e < E) unsafeAtomicAdd(&cnt[dst[e]], 1.0f);
}

__global__ __launch_bounds__(256)
void inv_kernel(float* __restrict__ cnt, int N) {
    int i = blockIdx.x * 256 + threadIdx.x;
    if (i < N) cnt[i] = 1.0f / fmaxf(cnt[i], 1.0f);
}

// -------------------- edge scatter: msg[dst] += xsrc[src] --------------------
// one thread = one edge x 4 consecutive channels
__global__ __launch_bounds__(256)
void scatter_kernel(const float* __restrict__ xsrc, const int* __restrict__ src,
                    const int* __restrict__ dst, float* __restrict__ msg, int E) {
    long long tid = (long long)blockIdx.x * 256 + threadIdx.x;
    if (tid >= (long long)E * 64) return;
    int e = (int)(tid >> 6);
    int c = (int)(tid & 63) << 2;
    int s = src[e], d = dst[e];
    const float4 v = *(const float4*)(xsrc + (size_t)s * HID + c);
    float* m = msg + (size_t)d * HID + c;
    unsafeAtomicAdd(m + 0, v.x);
    unsafeAtomicAdd(m + 1, v.y);
    unsafeAtomicAdd(m + 2, v.z);
    unsafeAtomicAdd(m + 3, v.w);
}

// -------------------- fused SAGE update (fp32 WMMA) --------------------
// out = relu( (msg .* inv[row]) @ Wl^T  +  xdst @ Wr^T  +  bias )
// One wave computes one 16x16 output tile; K=256 -> 64 V_WMMA_F32_16X16X4_F32
// per operand matrix, chained on a single v8f accumulator.
__global__ __launch_bounds__(256)
void sage_gemm_relu(const float* __restrict__ msg, const float* __restrict__ inv,
                    const float* __restrict__ xdst,
                    const float* __restrict__ Wl, const float* __restrict__ Wr,
                    const float* __restrict__ bias, float* __restrict__ out, int N) {
    const int lane = threadIdx.x & 31;
    const int wave = threadIdx.x >> 5;
    const int tile = blockIdx.x * 8 + wave;           // wave-uniform
    const int total = (N >> 4) << 4;                  // (N/16)*16 col-tiles == N
    if (tile >= total) return;                        // wave-uniform exit: EXEC stays all-1s
    const int it = tile >> 4;                         // row tile
    const int jt = tile & 15;                         // col tile (H/16 == 16)

    // A fragment addressing (16x4 fp32): lanes 0-15 -> K{0,1}, lanes 16-31 -> K{2,3}
    const int rA   = it * 16 + (lane & 15);
    const int kOff = (lane >> 4) << 1;                // 0 or 2
    const int cB   = jt * 16 + (lane & 15);           // B fragment: lane = N column
    const float invA = inv[rA];

    const float* aRow0 = msg  + (size_t)rA * HID + kOff;
    const float* bRow0 = Wl   + (size_t)cB * HID + kOff;   // B[k][n] = Wl[n][k]
    const float* aRow1 = xdst + (size_t)rA * HID + kOff;
    const float* bRow1 = Wr   + (size_t)cB * HID + kOff;

    v8f acc = {};
#pragma unroll 4
    for (int k0 = 0; k0 < HID; k0 += 4) {
        v2f a, b;
        a.x = aRow0[k0]     * invA;
        a.y = aRow0[k0 + 1] * invA;
        b.x = bRow0[k0];
        b.y = bRow0[k0 + 1];
        acc = __builtin_amdgcn_wmma_f32_16x16x4_f32(false, a, false, b,
                                                    (short)0, acc, false, false);
    }
#pragma unroll 4
    for (int k0 = 0; k0 < HID; k0 += 4) {
        v2f a, b;
        a.x = aRow1[k0];
        a.y = aRow1[k0 + 1];
        b.x = bRow1[k0];
        b.y = bRow1[k0 + 1];
        acc = __builtin_amdgcn_wmma_f32_16x16x4_f32(false, a, false, b,
                                                    (short)0, acc, false, false);
    }

    // C/D layout: VGPR r -> row it*16 + r (+8 for lanes 16-31), col = jt*16 + (lane&15)
    const float bv = bias[cB];
    const int rowBase = it * 16 + ((lane >> 4) << 3);
#pragma unroll
    for (int r = 0; r < 8; ++r) {
        float v = acc[r] + bv;
        v = v > 0.0f ? v : 0.0f;
        out[(size_t)(rowBase + r) * HID + cB] = v;
    }
}

// -------------------- column mean: out[c] += scale * sum_r x[r,c] --------------------
__global__ __launch_bounds__(256)
void colmean_kernel(const float* __restrict__ x, float* __restrict__ out,
                    int N, float scale) {
    const int c = threadIdx.x;                        // 256 columns
    float acc = 0.0f;
    for (int r = blockIdx.x; r < N; r += gridDim.x)
        acc += x[(size_t)r * HID + c];
    unsafeAtomicAdd(&out[c], acc * scale);
}

// -------------------- readout MLP: 512 -> 256 (relu) -> 2 --------------------
__global__ __launch_bounds__(256)
void head_kernel(const float* __restrict__ gr, const float* __restrict__ Wc1,
                 const float* __restrict__ bc1, const float* __restrict__ Wc2,
                 const float* __restrict__ bc2, float* __restrict__ out) {
    __shared__ float sg[2 * HID];
    __shared__ float sh[HID];
    const int t = threadIdx.x;
    sg[t]       = gr[t];
    sg[t + HID] = gr[t + HID];
    __syncthreads();
    float acc = bc1[t];
    for (int k = 0; k < 2 * HID; ++k) acc += sg[k] * Wc1[(size_t)t * 2 * HID + k];
    sh[t] = acc > 0.0f ? acc : 0.0f;
    __syncthreads();
    if (t < 2) {
        float a = bc2[t];
        for (int k = 0; k < HID; ++k) a += sh[k] * Wc2[(size_t)t * HID + k];
        out[t] = a;
    }
}

extern "C" void kernel_launch(void* const* d_in, const int* in_sizes, int n_in,
                              void* d_out, int out_size, void* d_ws, size_t ws_size,
                              hipStream_t stream) {
    const float* place_x = (const float*)d_in[0];
    const float* trans_x = (const float*)d_in[1];
    const int*   e_pre   = (const int*)d_in[2];
    const int*   e_post  = (const int*)d_in[3];
    const float* W_place = (const float*)d_in[4];
    const float* b_place = (const float*)d_in[5];
    const float* W_trans = (const float*)d_in[6];
    const float* b_trans = (const float*)d_in[7];
    const float* Wl_pre  = (const float*)d_in[8];
    const float* bl_pre  = (const float*)d_in[9];
    const float* Wr_pre  = (const float*)d_in[10];
    const float* Wl_post = (const float*)d_in[11];
    const float* bl_post = (const float*)d_in[12];
    const float* Wr_post = (const float*)d_in[13];
    const float* Wc1     = (const float*)d_in[14];
    const float* bc1     = (const float*)d_in[15];
    const float* Wc2     = (const float*)d_in[16];
    const float* bc2     = (const float*)d_in[17];

    const int P  = in_sizes[0];
    const int T  = in_sizes[1];
    const int Ea = in_sizes[2] / 2;   // edge_pre  count
    const int Eb = in_sizes[3] / 2;   // edge_post count
    const int* pre_src  = e_pre;            // place ids
    const int* pre_dst  = e_pre + Ea;       // transition ids
    const int* post_src = e_post;           // transition ids
    const int* post_dst = e_post + Eb;      // place ids

    const size_t NHp = (size_t)P * HID;
    const size_t NHt = (size_t)T * HID;

    float* w      = (float*)d_ws;
    float* bufP   = w;                         // p features (ping)
    float* bufT   = bufP + NHp;                // t features (ping)
    float* msgT   = bufT + NHt;                // edge sums into transitions
    float* msgP   = msgT + NHt;                // edge sums into places
    float* tN     = msgP + NHp;                // t features (pong)
    float* pN     = tN   + NHt;                // p features (pong)
    float* invT   = pN   + NHp;                // 1/max(deg,1) per transition
    float* invP   = invT + T;
    float* colsum = invP + P;                  // 512 graph-repr floats

    const int nb = 256;
    auto cdiv = [](long long a, long long b) { return (int)((a + b - 1) / b); };

    // ---- degrees (layer-invariant) ----
    hipMemsetAsync(invT, 0, sizeof(float) * (T + P + 512), stream);
    count_kernel<<<cdiv(Ea, nb), nb, 0, stream>>>(pre_dst, invT, Ea);
    count_kernel<<<cdiv(Eb, nb), nb, 0, stream>>>(post_dst, invP, Eb);
    inv_kernel<<<cdiv(T, nb), nb, 0, stream>>>(invT, T);
    inv_kernel<<<cdiv(P, nb), nb, 0, stream>>>(invP, P);

    // ---- input encoders ----
    encode_kernel<<<cdiv((long long)NHp, nb), nb, 0, stream>>>(place_x, W_place, b_place, bufP, P);
    encode_kernel<<<cdiv((long long)NHt, nb), nb, 0, stream>>>(trans_x, W_trans, b_trans, bufT, T);

    float* p_cur = bufP; float* t_cur = bufT;
    float* p_out = pN;   float* t_out = tN;

    for (int l = 0; l < 2; ++l) {
        const float* WlA = Wl_pre  + (size_t)l * HID * HID;
        const float* WrA = Wr_pre  + (size_t)l * HID * HID;
        const float* blA = bl_pre  + (size_t)l * HID;
        const float* WlB = Wl_post + (size_t)l * HID * HID;
        const float* WrB = Wr_post + (size_t)l * HID * HID;
        const float* blB = bl_post + (size_t)l * HID;

        // t_new = relu( mean_pre(p) @ WlA^T + t @ WrA^T + blA )
        hipMemsetAsync(msgT, 0, sizeof(float) * NHt, stream);
        scatter_kernel<<<cdiv((long long)Ea * 64, nb), nb, 0, stream>>>(p_cur, pre_src, pre_dst, msgT, Ea);
        sage_gemm_relu<<<cdiv(T, 8), nb, 0, stream>>>(msgT, invT, t_cur, WlA, WrA, blA, t_out, T);

        // p_new = relu( mean_post(t) @ WlB^T + p @ WrB^T + blB )
        hipMemsetAsync(msgP, 0, sizeof(float) * NHp, stream);
        scatter_kernel<<<cdiv((long long)Eb * 64, nb), nb, 0, stream>>>(t_cur, post_src, post_dst, msgP, Eb);
        sage_gemm_relu<<<cdiv(P, 8), nb, 0, stream>>>(msgP, invP, p_cur, WlB, WrB, blB, p_out, P);

        // ping-pong
        float* tp = p_cur; p_cur = p_out; p_out = tp;
        float* tt = t_cur; t_cur = t_out; t_out = tt;
    }

    // ---- readout ----
    colmean_kernel<<<256, nb, 0, stream>>>(p_cur, colsum,       P, 1.0f / (float)P);
    colmean_kernel<<<256, nb, 0, stream>>>(t_cur, colsum + HID, T, 1.0f / (float)T);
    head_kernel<<<1, nb, 0, stream>>>(colsum, Wc1, bc1, Wc2, bc2, (float*)d_out);
}